// CSAG_72267119722968
// MI455X (gfx1250) — compile-verified
//
#include <hip/hip_runtime.h>
#include <math.h>

#define B_SZ 32
#define DIM  256
#define HW   16384      // 128*128
#define EPS  1e-5f

typedef __attribute__((ext_vector_type(2))) float v2f;
typedef __attribute__((ext_vector_type(4))) float v4f;
typedef __attribute__((ext_vector_type(8))) float v8f;

// ---------------------------------------------------------------------------
// Kernel 1: global avg + max pool over each (b, d) plane of 16384 floats.
// One block per plane; coalesced 128-bit streaming loads with NON-TEMPORAL
// hint (x is 512 MB read exactly once -> do not thrash the 192 MB L2),
// then LDS tree reduction.
// ---------------------------------------------------------------------------
__global__ void CSAG_pool_kernel(const float* __restrict__ x,
                                 float* __restrict__ avg,
                                 float* __restrict__ mx) {
    const int p = blockIdx.x;                          // 0..8191 = b*DIM + d
    const v4f* xp = (const v4f*)(x + (size_t)p * HW);
    float s = 0.0f;
    float m = -INFINITY;
#pragma unroll
    for (int i = 0; i < 16; ++i) {
        v4f v = __builtin_nontemporal_load(&xp[i * 256 + threadIdx.x]);
        s += (v.x + v.y) + (v.z + v.w);
        m = fmaxf(m, fmaxf(fmaxf(v.x, v.y), fmaxf(v.z, v.w)));
    }
    __shared__ float ss[256];
    __shared__ float sm[256];
    ss[threadIdx.x] = s;
    sm[threadIdx.x] = m;
    __syncthreads();
    for (int off = 128; off > 0; off >>= 1) {
        if ((int)threadIdx.x < off) {
            ss[threadIdx.x] += ss[threadIdx.x + off];
            sm[threadIdx.x] = fmaxf(sm[threadIdx.x], sm[threadIdx.x + off]);
        }
        __syncthreads();
    }
    if (threadIdx.x == 0) {
        avg[p] = ss[0] * (1.0f / (float)HW);
        mx[p]  = sm[0];
    }
}

// ---------------------------------------------------------------------------
// Kernel 2: y = avg @ fc1_w^T + fc1_b + mx @ fc2_w^T + fc2_b  via
// V_WMMA_F32_16X16X4_F32. One workgroup (8 waves); 32 output tiles of 16x16;
// each wave owns 4 tiles and runs 128 WMMA k-steps per tile (K = 256 + 256).
//
// Fragment layouts (ISA 7.12.2, wave32):
//   A (16x4):  lane L holds A[L%16][2*(L/16)+j] in vgpr j
//   B (4x16):  lane L holds B[2*(L/16)+j][L%16] in vgpr j
//   D (16x16): lane L vgpr i holds D[i + 8*(L/16)][L%16]
// ---------------------------------------------------------------------------
__global__ void CSAG_gemm_wmma_kernel(const float* __restrict__ avg,
                                      const float* __restrict__ mx,
                                      const float* __restrict__ w1,
                                      const float* __restrict__ b1,
                                      const float* __restrict__ w2,
                                      const float* __restrict__ b2,
                                      float* __restrict__ y) {
    const int lane = (int)threadIdx.x & 31;
    const int wave = (int)threadIdx.x >> 5;   // 0..7
    const int g    = lane >> 4;               // half-wave: 0 or 1
    const int r    = lane & 15;               // 0..15

#pragma unroll
    for (int j = 0; j < 4; ++j) {
        const int t  = wave + 8 * j;          // tile id 0..31
        const int mt = t >> 4;                // 0..1   (batch tile)
        const int nt = t & 15;                // 0..15  (out-channel tile)

        v8f acc = {};

        // Pass 1: avg @ fc1_w^T   (B[k][n] = fc1_w[n][k])
        {
            const float* arow = avg + (size_t)(mt * 16 + r) * DIM;
            const float* brow = w1  + (size_t)(nt * 16 + r) * DIM;
            for (int kc = 0; kc < DIM; kc += 4) {
                const int k0 = kc + 2 * g;
                v2f a, bm;
                a.x  = arow[k0];     a.y  = arow[k0 + 1];
                bm.x = brow[k0];     bm.y = brow[k0 + 1];
                acc = __builtin_amdgcn_wmma_f32_16x16x4_f32(
                    false, a, false, bm, (short)0, acc, false, false);
            }
        }
        // Pass 2: mx @ fc2_w^T (logical K = 256..511)
        {
            const float* arow = mx + (size_t)(mt * 16 + r) * DIM;
            const float* brow = w2 + (size_t)(nt * 16 + r) * DIM;
            for (int kc = 0; kc < DIM; kc += 4) {
                const int k0 = kc + 2 * g;
                v2f a, bm;
                a.x  = arow[k0];     a.y  = arow[k0 + 1];
                bm.x = brow[k0];     bm.y = brow[k0 + 1];
                acc = __builtin_amdgcn_wmma_f32_16x16x4_f32(
                    false, a, false, bm, (short)0, acc, false, false);
            }
        }

        const int   col  = nt * 16 + r;       // output channel d
        const float bias = b1[col] + b2[col];
#pragma unroll
        for (int i = 0; i < 8; ++i) {
            const int row = mt * 16 + i + 8 * g;   // batch index
            y[(size_t)row * DIM + col] = acc[i] + bias;
        }
    }
}

// ---------------------------------------------------------------------------
// Kernel 3: GroupNorm(1 group) + depthwise q/k/v + 1-token attention +
// sigmoid gate. One block per batch, thread d = output channel.
// softmax row-max is exact: max_e(qs*k_e) = qs * (qs>=0 ? kmax : kmin).
// ---------------------------------------------------------------------------
__device__ __forceinline__ float block_sum(float v, float* red) {
    const int d = threadIdx.x;
    red[d] = v; __syncthreads();
    for (int off = 128; off > 0; off >>= 1) {
        if (d < off) red[d] += red[d + off];
        __syncthreads();
    }
    float r = red[0];
    __syncthreads();
    return r;
}
__device__ __forceinline__ float block_max(float v, float* red) {
    const int d = threadIdx.x;
    red[d] = v; __syncthreads();
    for (int off = 128; off > 0; off >>= 1) {
        if (d < off) red[d] = fmaxf(red[d], red[d + off]);
        __syncthreads();
    }
    float r = red[0];
    __syncthreads();
    return r;
}
__device__ __forceinline__ float block_min(float v, float* red) {
    const int d = threadIdx.x;
    red[d] = v; __syncthreads();
    for (int off = 128; off > 0; off >>= 1) {
        if (d < off) red[d] = fminf(red[d], red[d + off]);
        __syncthreads();
    }
    float r = red[0];
    __syncthreads();
    return r;
}

__global__ void CSAG_attn_kernel(const float* __restrict__ y,
                                 const float* __restrict__ nw,
                                 const float* __restrict__ nb,
                                 const float* __restrict__ qw,
                                 const float* __restrict__ kw,
                                 const float* __restrict__ vw,
                                 float* __restrict__ out) {
    const int b = blockIdx.x;
    const int d = threadIdx.x;

    __shared__ float red[256];
    __shared__ float ks[256];
    __shared__ float vs[256];

    const float yv = y[(size_t)b * DIM + d];

    // GroupNorm (one group over all channels, spatial 1x1; biased variance)
    const float mu  = block_sum(yv, red) * (1.0f / (float)DIM);
    const float dy  = yv - mu;
    const float var = block_sum(dy * dy, red) * (1.0f / (float)DIM);
    const float yn  = dy * rsqrtf(var + EPS) * nw[d] + nb[d];

    // depthwise 1x1 q/k/v
    const float q = yn * qw[d];
    ks[d] = yn * kw[d];
    vs[d] = yn * vw[d];
    __syncthreads();

    const float kmax = block_max(ks[d], red);
    const float kmin = block_min(ks[d], red);

    const float qs = q * 0.0625f;                    // HEAD_DIM^-0.5 = 1/16
    const float rowmax = qs * (qs >= 0.0f ? kmax : kmin);

    float den = 0.0f, num = 0.0f;
    for (int e = 0; e < DIM; ++e) {
        const float w = __expf(fmaf(qs, ks[e], -rowmax));
        den += w;
        num = fmaf(w, vs[e], num);
    }
    const float o = num / den;
    out[(size_t)b * DIM + d] = 1.0f / (1.0f + __expf(-o));
}

// ---------------------------------------------------------------------------
extern "C" void kernel_launch(void* const* d_in, const int* in_sizes, int n_in,
                              void* d_out, int out_size, void* d_ws, size_t ws_size,
                              hipStream_t stream) {
    const float* x      = (const float*)d_in[0];
    const float* fc1_w  = (const float*)d_in[1];
    const float* fc1_b  = (const float*)d_in[2];
    const float* fc2_w  = (const float*)d_in[3];
    const float* fc2_b  = (const float*)d_in[4];
    const float* norm_w = (const float*)d_in[5];
    const float* norm_b = (const float*)d_in[6];
    const float* q_w    = (const float*)d_in[7];
    const float* k_w    = (const float*)d_in[8];
    const float* v_w    = (const float*)d_in[9];

    float* avg = (float*)d_ws;               // 32*256
    float* mx  = avg + B_SZ * DIM;           // 32*256
    float* y   = mx  + B_SZ * DIM;           // 32*256

    CSAG_pool_kernel<<<B_SZ * DIM, 256, 0, stream>>>(x, avg, mx);
    CSAG_gemm_wmma_kernel<<<1, 256, 0, stream>>>(avg, mx, fc1_w, fc1_b,
                                                 fc2_w, fc2_b, y);
    CSAG_attn_kernel<<<B_SZ, 256, 0, stream>>>(y, norm_w, norm_b,
                                               q_w, k_w, v_w, (float*)d_out);
}